// MultiViewHyperConvLayer_58196806861293
// MI455X (gfx1250) — compile-verified
//
#include <hip/hip_runtime.h>

// MultiViewHyperConvLayer: out = PU_coo @ (UP_coo @ pois_embs)
// Memory/atomic-bound COO segment-sum SpMM x2 (AI ~ 0.25 FLOP/byte).
// Dense operands (51MB / 25.6MB) fit in the 192MB L2 -> gathers are L2-hot;
// scatter uses hw global_atomic_add_f32. SpMM#2 runs scale+accumulate on the
// otherwise-idle matrix pipe: D = diag(vals16) x rowChunk via 4 chained
// V_WMMA_F32_16X16X4_F32 (f32 -> same math as the reference).
// Broadcasts are wave-uniform -> v_readlane into SGPRs (no LDS round trips);
// B gathers for a chunk are preloaded as one clause before the WMMA chain so
// only one load-latency exposure is paid per chunk.

#define N_POI  100000
#define N_USER 50000
#define EMB    128
#define ROWB   (EMB * 4)   // row pitch in bytes (512)

typedef __attribute__((ext_vector_type(2))) float v2f;
typedef __attribute__((ext_vector_type(8))) float v8f;

__device__ __forceinline__ float readlane_f(float x, int l) {
  return __int_as_float(__builtin_amdgcn_readlane(__float_as_int(x), l));
}

__global__ void zero_f4_kernel(float4* __restrict__ p, int n4) {
  int i = blockIdx.x * blockDim.x + threadIdx.x;
  const int stride = gridDim.x * blockDim.x;
  const float4 z = make_float4(0.f, 0.f, 0.f, 0.f);
  for (; i < n4; i += stride) p[i] = z;
}

// SpMM #1 (VALU path): per nonzero (r,c,v): out[r,:] += v * dense[c,:]
// One wave owns 32 nonzeros per outer iter; per nonzero the 32 lanes move the
// 128-float row cooperatively (one b128 load + 4 f32 atomics per lane).
// (r,c,v) of nonzero j are wave-uniform -> readlane to SGPRs; row base is SALU.
__global__ void spmm_atomic_kernel(const int* __restrict__ rows,
                                   const int* __restrict__ cols,
                                   const float* __restrict__ vals,
                                   const float* __restrict__ dense,
                                   float* __restrict__ out, int nnz) {
  const int lane  = threadIdx.x & 31;
  const int wave  = (blockIdx.x * blockDim.x + threadIdx.x) >> 5;
  const int waves = (gridDim.x * blockDim.x) >> 5;
  const char* dbase = (const char*)dense;
  char*       obase = (char*)out;
  const uint32_t colb = (uint32_t)lane * 16u;   // this lane's float4 in a row

  for (int base = wave * 32; base < nnz; base += waves * 32) {
    const int e = base + lane;
    int r = 0, c = 0; float v = 0.f;
    if (e < nnz) { r = rows[e]; c = cols[e]; v = vals[e]; }
    // warm L2 for this lane's upcoming gather row (global_prefetch_b8)
    __builtin_prefetch(dense + (size_t)c * EMB, 0, 3);

    if (base + 32 <= nnz) {                     // fast path: full tile
      #pragma unroll 4
      for (int j = 0; j < 32; ++j) {
        const uint32_t cj = (uint32_t)__builtin_amdgcn_readlane(c, j);
        const uint32_t rj = (uint32_t)__builtin_amdgcn_readlane(r, j);
        const float    vj = readlane_f(v, j);
        const float4 d = *(const float4*)(dbase + cj * ROWB + colb);
        float* o = (float*)(obase + rj * ROWB + colb);
        unsafeAtomicAdd(o + 0, d.x * vj);
        unsafeAtomicAdd(o + 1, d.y * vj);
        unsafeAtomicAdd(o + 2, d.z * vj);
        unsafeAtomicAdd(o + 3, d.w * vj);
      }
    } else {                                    // tail tile
      const int cnt = nnz - base;
      for (int j = 0; j < cnt; ++j) {
        const uint32_t cj = (uint32_t)__builtin_amdgcn_readlane(c, j);
        const uint32_t rj = (uint32_t)__builtin_amdgcn_readlane(r, j);
        const float    vj = readlane_f(v, j);
        const float4 d = *(const float4*)(dbase + cj * ROWB + colb);
        float* o = (float*)(obase + rj * ROWB + colb);
        unsafeAtomicAdd(o + 0, d.x * vj);
        unsafeAtomicAdd(o + 1, d.y * vj);
        unsafeAtomicAdd(o + 2, d.z * vj);
        unsafeAtomicAdd(o + 3, d.w * vj);
      }
    }
  }
}

// SpMM #2 (matrix-pipe path): 16 nonzeros per wave iter.
//   D(16x16) = diag(v0..v15) x B,  B[k][n] = dense[cols[k]][cb+n]
// as 4 chained V_WMMA_F32_16X16X4_F32 per 16-column chunk (8 chunks).
// A 16x4 layout: lane l holds M=l&15; VGPR j holds K = 4s + 2*(l>>4) + j.
// B 4x16 layout: VGPR j, half h holds row K = 4s + j + 2h, column N = l&15.
// C/D layout: VGPR j holds M = j + 8*(l>>4), N = l&15 -> direct atomic scatter.
// Broadcasts + row offsets hoisted out of the chunk loop (chunk-invariant);
// per chunk: 8-load clause -> 4-WMMA chain -> 8-atomic clause.
__global__ void spmm_wmma_kernel(const int* __restrict__ rows,
                                 const int* __restrict__ cols,
                                 const float* __restrict__ vals,
                                 const float* __restrict__ dense,
                                 float* __restrict__ out, int nnz) {
  const int lane  = threadIdx.x & 31;
  const int half  = lane >> 4;
  const int m     = lane & 15;
  const int wave  = (blockIdx.x * blockDim.x + threadIdx.x) >> 5;
  const int waves = (gridDim.x * blockDim.x) >> 5;
  const char* dbase = (const char*)dense;
  char*       obase = (char*)out;

  for (int base = wave * 16; base < nnz; base += waves * 16) {
    const int e = base + m;                     // lanes 16..31 mirror 0..15
    int r = 0, c = 0; float v = 0.f;
    if (e < nnz) { r = rows[e]; c = cols[e]; v = vals[e]; }
    __builtin_prefetch(dense + (size_t)c * EMB, 0, 3);

    // Hoisted wave-uniform broadcasts (v_readlane -> SGPR) + per-half select.
    uint32_t rowoff0[4], rowoff1[4];            // B source-row byte offsets
    v2f      amat[4];                           // diagonal A slices
    #pragma unroll
    for (int s = 0; s < 4; ++s) {
      const int k0 = 4 * s + 2 * half, k1 = k0 + 1;
      const int cA0 = __builtin_amdgcn_readlane(c, 4 * s + 0);
      const int cA1 = __builtin_amdgcn_readlane(c, 4 * s + 1);
      const int cB0 = __builtin_amdgcn_readlane(c, 4 * s + 2);
      const int cB1 = __builtin_amdgcn_readlane(c, 4 * s + 3);
      rowoff0[s] = (uint32_t)(half ? cB0 : cA0) * (uint32_t)ROWB;
      rowoff1[s] = (uint32_t)(half ? cB1 : cA1) * (uint32_t)ROWB;
      amat[s].x = (m == k0) ? v : 0.f;          // A[m][k] = v_m iff m==k
      amat[s].y = (m == k1) ? v : 0.f;
    }
    uint32_t outoff[8];                         // D target-row byte offsets
    #pragma unroll
    for (int j = 0; j < 8; ++j) {
      const int rA = __builtin_amdgcn_readlane(r, j);
      const int rB = __builtin_amdgcn_readlane(r, j + 8);
      outoff[j] = (uint32_t)(half ? rB : rA) * (uint32_t)ROWB;
    }

    for (int cb = 0; cb < EMB; cb += 16) {
      const uint32_t colb = (uint32_t)(cb + m) * 4u;
      // Preload all B operands for this chunk: one 8-load clause, one wait.
      v2f bmat[4];
      #pragma unroll
      for (int s = 0; s < 4; ++s) {
        bmat[s].x = *(const float*)(dbase + rowoff0[s] + colb);
        bmat[s].y = *(const float*)(dbase + rowoff1[s] + colb);
      }
      // Chained accumulation on the matrix pipe.
      v8f acc = {};
      #pragma unroll
      for (int s = 0; s < 4; ++s)
        acc = __builtin_amdgcn_wmma_f32_16x16x4_f32(
            /*neg_a=*/false, amat[s], /*neg_b=*/false, bmat[s],
            /*c_mod=*/(short)0, acc, /*reuse_a=*/false, /*reuse_b=*/false);
      // Scatter D: lane holds rows M = j + 8*half at column cb + m.
      #pragma unroll
      for (int j = 0; j < 8; ++j)
        unsafeAtomicAdd((float*)(obase + outoff[j] + colb), acc[j]);
    }
  }
}

extern "C" void kernel_launch(void* const* d_in, const int* in_sizes, int n_in,
                              void* d_out, int out_size, void* d_ws, size_t ws_size,
                              hipStream_t stream) {
  // setup_inputs() order:
  // 0: pois_embs f32 [N_POI*EMB]   1: pad_all_train_sessions (unused, i64)
  // 2: up_rows i32  3: up_cols i32  4: up_vals f32
  // 5: pu_rows i32  6: pu_cols i32  7: pu_vals f32
  const float* pois    = (const float*)d_in[0];
  const int*   up_rows = (const int*)d_in[2];
  const int*   up_cols = (const int*)d_in[3];
  const float* up_vals = (const float*)d_in[4];
  const int*   pu_rows = (const int*)d_in[5];
  const int*   pu_cols = (const int*)d_in[6];
  const float* pu_vals = (const float*)d_in[7];
  float* out = (float*)d_out;
  float* msg = (float*)d_ws;                    // [N_USER, EMB] = 25.6 MB scratch

  const int nnz_up = in_sizes[2];
  const int nnz_pu = in_sizes[5];

  zero_f4_kernel<<<1024, 256, 0, stream>>>((float4*)msg, N_USER * EMB / 4);
  spmm_atomic_kernel<<<2048, 256, 0, stream>>>(up_rows, up_cols, up_vals,
                                               pois, msg, nnz_up);
  zero_f4_kernel<<<2048, 256, 0, stream>>>((float4*)out, N_POI * EMB / 4);
  spmm_wmma_kernel<<<2048, 256, 0, stream>>>(pu_rows, pu_cols, pu_vals,
                                             msg, out, nnz_pu);
}